// GATTest_47588237639832
// MI455X (gfx1250) — compile-verified
//
#include <hip/hip_runtime.h>
#include <math.h>

// ---------------------------------------------------------------------------
// GAT 2-layer forward, but only the dependency cone of node `mask_idx`.
// Shapes: N=8192 nodes, E=32768 edges (+N self-loops), C=768, H1=8, OUT=768.
// Output = [1,2] fp32.
// ---------------------------------------------------------------------------

#define FEAT   768
#define HEADS  8
#define HF     (HEADS * FEAT)   // 6144
#define S1C    256              // cap: nodes whose layer-1 output is needed
#define S2C    512              // cap: nodes whose x@W1 row is needed
#define MAXE   512              // cap: in-edges per collected node

typedef float v2f __attribute__((ext_vector_type(2)));
typedef float v8f __attribute__((ext_vector_type(8)));

// ---------------------------------------------------------------------------
// Utility kernels
// ---------------------------------------------------------------------------
__global__ void k_zero_i32(int* p, int n) {
    int i = blockIdx.x * blockDim.x + threadIdx.x;
    if (i < n) p[i] = 0;
}

// Mark sources of edges whose dst == *maskPtr (self-loop edge E+m marks m itself).
__global__ void k_mark_dst_eq(const int* __restrict__ ei, int E, int N,
                              const int* __restrict__ maskPtr, int* __restrict__ mark) {
    int e = blockIdx.x * blockDim.x + threadIdx.x;
    int ET = E + N;
    if (e >= ET) return;
    int m   = *maskPtr;
    int dst = (e < E) ? ei[E + e] : (e - E);
    if (dst == m) {
        int src = (e < E) ? ei[e] : (e - E);
        mark[src] = 1;
    }
}

// Mark sources of edges whose dst is marked in mark1 (self-loops propagate S1 into S2).
__global__ void k_mark_dst_in(const int* __restrict__ ei, int E, int N,
                              const int* __restrict__ mark1, int* __restrict__ mark2) {
    int e = blockIdx.x * blockDim.x + threadIdx.x;
    int ET = E + N;
    if (e >= ET) return;
    int dst = (e < E) ? ei[E + e] : (e - E);
    if (mark1[dst]) {
        int src = (e < E) ? ei[e] : (e - E);
        mark2[src] = 1;
    }
}

// Deterministic (node-ordered) single-block compaction of a 0/1 mark array.
// Writes list[idx]=node, map[node]=idx, *cnt = count (clamped to cap).
__global__ __launch_bounds__(256) void k_compact(const int* __restrict__ mark, int N,
                                                 int* __restrict__ list, int* __restrict__ map,
                                                 int* __restrict__ cnt, int cap) {
    __shared__ int wcnt[8];
    __shared__ int total;
    int tid = threadIdx.x, lane = tid & 31, wv = tid >> 5;
    if (tid == 0) total = 0;
    __syncthreads();
    for (int base = 0; base < N; base += 256) {
        int node = base + tid;
        bool pred = (node < N) && (mark[node] != 0);
        unsigned msk = __builtin_amdgcn_ballot_w32(pred);
        int lpre = __popc(msk & ((1u << lane) - 1u));
        if (lane == 0) wcnt[wv] = __popc(msk);
        __syncthreads();
        int off = total;
        for (int w = 0; w < wv; ++w) off += wcnt[w];
        if (pred) {
            int idx = off + lpre;
            if (idx < cap) { list[idx] = node; map[node] = idx; }
        }
        int csum = 0;
        for (int w = 0; w < 8; ++w) csum += wcnt[w];
        __syncthreads();
        if (tid == 0) total = min(total + csum, cap);
        __syncthreads();
    }
    if (tid == 0) *cnt = total;
}

// ---------------------------------------------------------------------------
// WMMA fp32 GEMM: C[r, 0..N) = sum_k A[row(r), k] * B[k, n]
//   One wave computes a 16x64 tile = 4 x (16x16) via V_WMMA_F32_16X16X4_F32,
//   so one A-fragment (b64 load) feeds 4 WMMAs per K-step of 4.
//   Rows >= *nRowsPtr are clamped (their C rows are never consumed downstream;
//   clamping keeps all addresses in-bounds and the wave divergence-free).
//   A 16x4 layout : lanes 0-15 hold (K=k0,k0+1), lanes 16-31 hold (K=k0+2,k0+3).
//   B 4x16 layout : v0 = rows k0 / k0+2 striped over lane halves, v1 = k0+1 / k0+3.
//   D layout      : VGPR i -> row i (lanes 0-15) / row 8+i (lanes 16-31), col = lane&15.
// ---------------------------------------------------------------------------
__global__ __launch_bounds__(32) void wmma_gemm_f32(
    const float* __restrict__ A, const int* __restrict__ rowList,
    const int* __restrict__ nRowsPtr, const float* __restrict__ B,
    float* __restrict__ C, int K, int N, int lda) {
    int nRows  = *nRowsPtr;
    int mTiles = (nRows + 15) >> 4;
    int tm = blockIdx.y, tn = blockIdx.x;
    if (tm >= mTiles) return;

    int lane   = threadIdx.x;
    int mloc   = lane & 15;
    int kOff   = (lane >> 4) * 2;                 // 0 for lanes 0-15, 2 for 16-31
    int mclamp = min(tm * 16 + mloc, nRows - 1);  // divergence-free, in-bounds
    long arow  = rowList ? (long)rowList[mclamp] : (long)mclamp;

    const float* aPtr = A + arow * (long)lda + kOff;
    const float* bPtr = B + (long)kOff * N + tn * 64 + mloc;
    long bStep = 4L * (long)N;

    v8f acc0 = {0.f,0.f,0.f,0.f,0.f,0.f,0.f,0.f};
    v8f acc1 = acc0, acc2 = acc0, acc3 = acc0;

#pragma unroll 4
    for (int k0 = 0; k0 < K; k0 += 4) {
        v2f a = *(const v2f*)aPtr;                // K=k0+kOff, k0+kOff+1
        const float* b0 = bPtr;                   // row k0+kOff
        const float* b1 = bPtr + N;               // row k0+kOff+1
        v2f f0, f1, f2, f3;
        f0.x = b0[0];  f0.y = b1[0];
        f1.x = b0[16]; f1.y = b1[16];
        f2.x = b0[32]; f2.y = b1[32];
        f3.x = b0[48]; f3.y = b1[48];
        acc0 = __builtin_amdgcn_wmma_f32_16x16x4_f32(false, a, false, f0, (short)0, acc0, false, false);
        acc1 = __builtin_amdgcn_wmma_f32_16x16x4_f32(false, a, false, f1, (short)0, acc1, false, false);
        acc2 = __builtin_amdgcn_wmma_f32_16x16x4_f32(false, a, false, f2, (short)0, acc2, false, false);
        acc3 = __builtin_amdgcn_wmma_f32_16x16x4_f32(false, a, false, f3, (short)0, acc3, false, false);
        aPtr += 4;
        bPtr += bStep;
    }

    union { v8f v; float f[8]; } u0, u1, u2, u3;
    u0.v = acc0; u1.v = acc1; u2.v = acc2; u3.v = acc3;
    int rbase = tm * 16 + (lane >> 4) * 8;
    float* cBase = C + (long)rbase * N + tn * 64 + mloc;
#pragma unroll
    for (int i = 0; i < 8; ++i) {
        long ro = (long)i * N;
        cBase[ro +  0] = u0.f[i];
        cBase[ro + 16] = u1.f[i];
        cBase[ro + 32] = u2.f[i];
        cBase[ro + 48] = u3.f[i];
    }
}

// ---------------------------------------------------------------------------
// Per-(row, head) attention dot products: aS = <H_row_h, attS_h>, aD likewise.
// ---------------------------------------------------------------------------
__global__ __launch_bounds__(32) void attn_dots(
    const float* __restrict__ H, const int* __restrict__ nPtr,
    const float* __restrict__ attS, const float* __restrict__ attD,
    int heads, float* __restrict__ aS, float* __restrict__ aD) {
    int r = blockIdx.x, h = blockIdx.y;
    if (r >= *nPtr) return;
    int lane = threadIdx.x;
    const float* row = H + (long)r * heads * FEAT + (long)h * FEAT;
    const float* s   = attS + (long)h * FEAT;
    const float* dv  = attD + (long)h * FEAT;
    float accS = 0.f, accD = 0.f;
    for (int c = lane; c < FEAT; c += 32) {
        float v = row[c];
        accS += v * s[c];
        accD += v * dv[c];
    }
    for (int off = 16; off > 0; off >>= 1) {
        accS += __shfl_down(accS, off, 32);
        accD += __shfl_down(accD, off, 32);
    }
    if (lane == 0) { aS[r * heads + h] = accS; aD[r * heads + h] = accD; }
}

// ---------------------------------------------------------------------------
// Layer-1 segment softmax + aggregation + bias + ELU for each node in S1.
// One block per S1 row; edges collected in edge-index order (deterministic).
// ---------------------------------------------------------------------------
__global__ __launch_bounds__(256) void k_layer1_agg(
    const int* __restrict__ ei, int E, int N,
    const int* __restrict__ list1, const int* __restrict__ map2,
    const int* __restrict__ counts,
    const float* __restrict__ aS1, const float* __restrict__ aD1,
    const float* __restrict__ H1, const float* __restrict__ b1,
    float* __restrict__ h2in) {
    int n1 = counts[0];
    int jr = blockIdx.x;
    if (jr >= n1) return;
    int j  = list1[jr];
    int rj = map2[j];

    __shared__ int   eRow[MAXE];
    __shared__ float eAlpha[MAXE * HEADS];
    __shared__ float shMax[HEADS], shDen[HEADS];
    __shared__ int   wcnt[8];
    __shared__ int   total;

    int tid = threadIdx.x, lane = tid & 31, wv = tid >> 5;
    if (tid == 0) total = 0;
    __syncthreads();

    int ET = E + N;
    for (int base = 0; base < ET; base += 256) {
        int e = base + tid;
        bool pred = false; int srow = 0;
        if (e < ET) {
            int dst = (e < E) ? ei[E + e] : (e - E);
            if (dst == j) {
                int src = (e < E) ? ei[e] : (e - E);
                srow = map2[src];
                pred = true;
            }
        }
        unsigned msk = __builtin_amdgcn_ballot_w32(pred);
        int lpre = __popc(msk & ((1u << lane) - 1u));
        if (lane == 0) wcnt[wv] = __popc(msk);
        __syncthreads();
        int off = total;
        for (int w = 0; w < wv; ++w) off += wcnt[w];
        if (pred) { int idx = off + lpre; if (idx < MAXE) eRow[idx] = srow; }
        int csum = 0;
        for (int w = 0; w < 8; ++w) csum += wcnt[w];
        __syncthreads();
        if (tid == 0) total = min(total + csum, MAXE);
        __syncthreads();
    }
    int nE = total;

    if (tid < HEADS) {
        int h = tid;
        float ad = aD1[rj * HEADS + h];
        float mx = -3.4e38f;
        for (int e = 0; e < nE; ++e) {
            float l = aS1[eRow[e] * HEADS + h] + ad;
            l = (l > 0.f) ? l : 0.2f * l;
            mx = fmaxf(mx, l);
        }
        float s = 0.f;
        for (int e = 0; e < nE; ++e) {
            float l = aS1[eRow[e] * HEADS + h] + ad;
            l = (l > 0.f) ? l : 0.2f * l;
            s += expf(l - mx);
        }
        shMax[h] = mx;
        shDen[h] = s + 1e-16f;
    }
    __syncthreads();

    for (int idx = tid; idx < nE * HEADS; idx += 256) {
        int e = idx >> 3, h = idx & 7;
        float l = aS1[eRow[e] * HEADS + h] + aD1[rj * HEADS + h];
        l = (l > 0.f) ? l : 0.2f * l;
        eAlpha[idx] = expf(l - shMax[h]) / shDen[h];
    }
    __syncthreads();

    for (int c = tid; c < HF; c += 256) {
        int h = c / FEAT;
        float acc = 0.f;
        for (int e = 0; e < nE; ++e)
            acc += eAlpha[e * HEADS + h] * H1[(long)eRow[e] * HF + c];
        float v = acc + b1[c];
        h2in[(long)jr * HF + c] = (v > 0.f) ? v : (expf(v) - 1.f);  // ELU
    }
}

// ---------------------------------------------------------------------------
// Layer-2 softmax/aggregate at node m, then fc (1536->768) and cls (768->2).
// Single block; all reductions in canonical order (deterministic).
// ---------------------------------------------------------------------------
__global__ __launch_bounds__(256) void k_final(
    const int* __restrict__ ei, int E, int N,
    const int* __restrict__ maskPtr, const int* __restrict__ map1,
    const float* __restrict__ aS2, const float* __restrict__ aD2,
    const float* __restrict__ G, const float* __restrict__ b2,
    const float* __restrict__ x,
    const float* __restrict__ fcw, const float* __restrict__ fcb,
    const float* __restrict__ clw, const float* __restrict__ clb,
    float* __restrict__ out) {
    __shared__ int   eR[MAXE];
    __shared__ float alpha[MAXE];
    __shared__ float cat[2 * FEAT];
    __shared__ float fco[FEAT];
    __shared__ float part0[256], part1[256];
    __shared__ int   wcnt[8];
    __shared__ int   total;
    __shared__ float shMax, shDen;

    int m = *maskPtr;
    int tid = threadIdx.x, lane = tid & 31, wv = tid >> 5;
    if (tid == 0) total = 0;
    __syncthreads();

    int ET = E + N;
    for (int base = 0; base < ET; base += 256) {
        int e = base + tid;
        bool pred = false; int srow = 0;
        if (e < ET) {
            int dst = (e < E) ? ei[E + e] : (e - E);
            if (dst == m) {
                int src = (e < E) ? ei[e] : (e - E);
                srow = map1[src];
                pred = true;
            }
        }
        unsigned msk = __builtin_amdgcn_ballot_w32(pred);
        int lpre = __popc(msk & ((1u << lane) - 1u));
        if (lane == 0) wcnt[wv] = __popc(msk);
        __syncthreads();
        int off = total;
        for (int w = 0; w < wv; ++w) off += wcnt[w];
        if (pred) { int idx = off + lpre; if (idx < MAXE) eR[idx] = srow; }
        int csum = 0;
        for (int w = 0; w < 8; ++w) csum += wcnt[w];
        __syncthreads();
        if (tid == 0) total = min(total + csum, MAXE);
        __syncthreads();
    }
    int nE = total;
    int rm = map1[m];

    if (tid == 0) {
        float ad = aD2[rm];
        float mx = -3.4e38f;
        for (int e = 0; e < nE; ++e) {
            float l = aS2[eR[e]] + ad;
            l = (l > 0.f) ? l : 0.2f * l;
            mx = fmaxf(mx, l);
        }
        float s = 0.f;
        for (int e = 0; e < nE; ++e) {
            float l = aS2[eR[e]] + ad;
            l = (l > 0.f) ? l : 0.2f * l;
            s += expf(l - mx);
        }
        shMax = mx;
        shDen = s + 1e-16f;
    }
    __syncthreads();

    for (int e = tid; e < nE; e += 256) {
        float l = aS2[eR[e]] + aD2[rm];
        l = (l > 0.f) ? l : 0.2f * l;
        alpha[e] = expf(l - shMax) / shDen;
    }
    __syncthreads();

    for (int c = tid; c < FEAT; c += 256) {
        float acc = 0.f;
        for (int e = 0; e < nE; ++e)
            acc += alpha[e] * G[(long)eR[e] * FEAT + c];
        cat[c] = acc + b2[c];                       // x_gemb
        cat[FEAT + c] = x[(long)m * FEAT + c];      // x_emb
    }
    __syncthreads();

    for (int c = tid; c < FEAT; c += 256) {
        float acc = fcb[c];
        for (int k = 0; k < 2 * FEAT; ++k)
            acc += cat[k] * fcw[(long)k * FEAT + c];
        fco[c] = acc;
    }
    __syncthreads();

    float p0 = 0.f, p1 = 0.f;
    for (int c = tid; c < FEAT; c += 256) {
        float v = fco[c];
        p0 += v * clw[c * 2 + 0];
        p1 += v * clw[c * 2 + 1];
    }
    part0[tid] = p0; part1[tid] = p1;
    __syncthreads();
    if (tid == 0) {
        float s0 = clb[0], s1 = clb[1];
        for (int t = 0; t < 256; ++t) { s0 += part0[t]; s1 += part1[t]; }
        out[0] = s0;
        out[1] = s1;
    }
}

// ---------------------------------------------------------------------------
// Host launcher
// ---------------------------------------------------------------------------
extern "C" void kernel_launch(void* const* d_in, const int* in_sizes, int n_in,
                              void* d_out, int out_size, void* d_ws, size_t ws_size,
                              hipStream_t stream) {
    (void)n_in; (void)out_size; (void)ws_size;
    const float* x     = (const float*)d_in[0];
    const float* W1    = (const float*)d_in[1];
    const float* attS1 = (const float*)d_in[2];
    const float* attD1 = (const float*)d_in[3];
    const float* b1    = (const float*)d_in[4];
    const float* W2    = (const float*)d_in[5];
    const float* attS2 = (const float*)d_in[6];
    const float* attD2 = (const float*)d_in[7];
    const float* b2    = (const float*)d_in[8];
    const float* fcw   = (const float*)d_in[9];
    const float* fcb   = (const float*)d_in[10];
    const float* clw   = (const float*)d_in[11];
    const float* clb   = (const float*)d_in[12];
    const int*   ei    = (const int*)d_in[13];
    const int*   maskP = (const int*)d_in[14];
    float* out = (float*)d_out;

    int E = in_sizes[13] / 2;
    int N = in_sizes[0] / FEAT;
    int ET = E + N;

    char* ws = (char*)d_ws;
    size_t off = 0;
    auto alloc = [&](size_t bytes) -> void* {
        void* p = ws + off;
        off += (bytes + 255) & ~(size_t)255;
        return p;
    };
    int*   mark1  = (int*)alloc((size_t)N * 4);
    int*   mark2  = (int*)alloc((size_t)N * 4);
    int*   map1   = (int*)alloc((size_t)N * 4);
    int*   map2   = (int*)alloc((size_t)N * 4);
    int*   list1  = (int*)alloc((size_t)S1C * 4);
    int*   list2  = (int*)alloc((size_t)S2C * 4);
    int*   counts = (int*)alloc(2 * 4);
    float* H1     = (float*)alloc((size_t)S2C * HF * 4);   // x@W1 rows for S2
    float* aS1    = (float*)alloc((size_t)S2C * HEADS * 4);
    float* aD1    = (float*)alloc((size_t)S2C * HEADS * 4);
    float* h2in   = (float*)alloc((size_t)S1C * HF * 4);   // ELU(layer1) rows for S1
    float* G      = (float*)alloc((size_t)S1C * FEAT * 4); // h2in@W2 rows
    float* aS2    = (float*)alloc((size_t)S1C * 4);
    float* aD2    = (float*)alloc((size_t)S1C * 4);

    k_zero_i32<<<(N + 255) / 256, 256, 0, stream>>>(mark1, N);
    k_zero_i32<<<(N + 255) / 256, 256, 0, stream>>>(mark2, N);
    k_mark_dst_eq<<<(ET + 255) / 256, 256, 0, stream>>>(ei, E, N, maskP, mark1);
    k_compact<<<1, 256, 0, stream>>>(mark1, N, list1, map1, counts + 0, S1C);
    k_mark_dst_in<<<(ET + 255) / 256, 256, 0, stream>>>(ei, E, N, mark1, mark2);
    k_compact<<<1, 256, 0, stream>>>(mark2, N, list2, map2, counts + 1, S2C);

    // H1[r,:] = x[list2[r],:] @ W1   (K=768, N=6144)
    {
        dim3 grid(HF / 64, S2C / 16);
        wmma_gemm_f32<<<grid, 32, 0, stream>>>(x, list2, counts + 1, W1, H1,
                                               FEAT, HF, FEAT);
    }
    attn_dots<<<dim3(S2C, HEADS), 32, 0, stream>>>(H1, counts + 1, attS1, attD1,
                                                   HEADS, aS1, aD1);
    k_layer1_agg<<<S1C, 256, 0, stream>>>(ei, E, N, list1, map2, counts,
                                          aS1, aD1, H1, b1, h2in);
    // G[r,:] = h2in[r,:] @ W2   (K=6144, N=768)
    {
        dim3 grid(FEAT / 64, S1C / 16);
        wmma_gemm_f32<<<grid, 32, 0, stream>>>(h2in, nullptr, counts + 0, W2, G,
                                               HF, FEAT, HF);
    }
    attn_dots<<<dim3(S1C, 1), 32, 0, stream>>>(G, counts + 0, attS2, attD2,
                                               1, aS2, aD2);
    k_final<<<1, 256, 0, stream>>>(ei, E, N, maskP, map1, aS2, aD2, G, b2, x,
                                   fcw, fcb, clw, clb, out);
}